// LrmcSeededPoolGCN_49615462204214
// MI455X (gfx1250) — compile-verified
//
#include <hip/hip_runtime.h>

typedef float v2f __attribute__((ext_vector_type(2)));
typedef float v8f __attribute__((ext_vector_type(8)));

#define KCLUST 1000
#define KPAD   1008   // KCLUST rounded up to multiple of 16

// ---------------- utility ----------------
__global__ void zero_f32(float* __restrict__ p, size_t n) {
    size_t i = (size_t)blockIdx.x * blockDim.x + threadIdx.x;
    size_t stride = (size_t)gridDim.x * blockDim.x;
    for (; i < n; i += stride) p[i] = 0.0f;
}

// per-edge: degree histogram on dst, cluster adjacency A[cv, cu] += 1
__global__ void edge_deg_A(const int* __restrict__ src, const int* __restrict__ dst,
                           const int* __restrict__ cid, float* __restrict__ deg,
                           float* __restrict__ Amat, int E) {
    int e = blockIdx.x * blockDim.x + threadIdx.x;
    if (e >= E) return;
    int s = src[e], d = dst[e];
    atomicAdd(&deg[d], 1.0f);
    atomicAdd(&Amat[(size_t)cid[d] * KCLUST + cid[s]], 1.0f);
}

// per-node: deg -> dinv in place (deg+1 for self loop), cluster counts
__global__ void node_prep(float* __restrict__ deg, const int* __restrict__ cid,
                          float* __restrict__ counts, int Nn) {
    int n = blockIdx.x * blockDim.x + threadIdx.x;
    if (n >= Nn) return;
    deg[n] = rsqrtf(deg[n] + 1.0f);
    atomicAdd(&counts[cid[n]], 1.0f);
}

// ---------------- fp32 WMMA GEMM, N fixed at 64: C[M,64] = A[M,Kd] @ B[Kd,64] --------
// One wave per 16-row strip: 4 accumulator tiles, A-fragment reused across the 4
// N-tiles.  Requires M % 16 == 0 (caller pads), Kd % 4 == 0.  EXEC all-ones.
__global__ void wmma_gemm_n64(const float* __restrict__ A, const float* __restrict__ B,
                              float* __restrict__ C, int M, int Kd) {
    int mt   = (int)((blockIdx.x * blockDim.x + threadIdx.x) >> 5);
    int lane = threadIdx.x & 31;
    if (mt * 16 >= M) return;                    // wave-uniform
    int row   = lane & 15;
    int khalf = lane >> 4;
    const float* ap = A + (size_t)(mt * 16 + row) * Kd + 2 * khalf;
    const float* bp = B + (size_t)(2 * khalf) * 64 + row;
    v8f acc0 = {}, acc1 = {}, acc2 = {}, acc3 = {};
    #pragma unroll 2
    for (int k0 = 0; k0 < Kd; k0 += 4) {
        v2f a = *(const v2f*)ap;
        v2f b0, b1, b2, b3;
        b0.x = bp[0];       b0.y = bp[64];
        b1.x = bp[16];      b1.y = bp[80];
        b2.x = bp[32];      b2.y = bp[96];
        b3.x = bp[48];      b3.y = bp[112];
        acc0 = __builtin_amdgcn_wmma_f32_16x16x4_f32(false, a, false, b0, (short)0, acc0, false, false);
        acc1 = __builtin_amdgcn_wmma_f32_16x16x4_f32(false, a, false, b1, (short)0, acc1, false, false);
        acc2 = __builtin_amdgcn_wmma_f32_16x16x4_f32(false, a, false, b2, (short)0, acc2, false, false);
        acc3 = __builtin_amdgcn_wmma_f32_16x16x4_f32(false, a, false, b3, (short)0, acc3, false, false);
        ap += 4;
        bp += 256;
    }
    float* cp = C + (size_t)(mt * 16 + 8 * khalf) * 64 + row;
    #pragma unroll
    for (int v = 0; v < 8; ++v) {
        cp[(size_t)v * 64]      = acc0[v];
        cp[(size_t)v * 64 + 16] = acc1[v];
        cp[(size_t)v * 64 + 32] = acc2[v];
        cp[(size_t)v * 64 + 48] = acc3[v];
    }
}

// ---------------- edge scatter: agg[dst] += xw[src] * dinv[src]*dinv[dst] ----------------
// one wave32 per edge, 2 channels per lane (64 channels)
__global__ void scatter_edges(const int* __restrict__ src, const int* __restrict__ dst,
                              const float* __restrict__ dinv, const float* __restrict__ xw,
                              float* __restrict__ agg, int E) {
    int wid = (int)((blockIdx.x * blockDim.x + threadIdx.x) >> 5);
    int lane = threadIdx.x & 31;
    if (wid >= E) return;
    int s = src[wid], d = dst[wid];
    float norm = dinv[s] * dinv[d];
    const float* xs = xw + (size_t)s * 64;
    float* ad = agg + (size_t)d * 64;
    int c = lane * 2;
    v2f v = *(const v2f*)(xs + c);
    atomicAdd(&ad[c],     v.x * norm);
    atomicAdd(&ad[c + 1], v.y * norm);
}

// ---------------- per-node: x1 = relu(agg + xw*dinv^2 + b1), gate = tanh(x1.score_w),
// pooled sums[cluster] += x1*gate.  one wave per node. ----------------
__global__ void x1_gate(const float* __restrict__ agg, const float* __restrict__ xw,
                        const float* __restrict__ dinv, const float* __restrict__ b1,
                        const float* __restrict__ score_w, const int* __restrict__ cid,
                        float* __restrict__ x1, float* __restrict__ sums, int Nn) {
    int wid = (int)((blockIdx.x * blockDim.x + threadIdx.x) >> 5);
    int lane = threadIdx.x & 31;
    if (wid >= Nn) return;
    float di = dinv[wid];
    float sl = di * di;
    size_t base = (size_t)wid * 64;
    int c = lane * 2;
    v2f ag = *(const v2f*)(agg + base + c);
    v2f xv = *(const v2f*)(xw + base + c);
    float v0 = fmaxf(ag.x + xv.x * sl + b1[c],     0.0f);
    float v1 = fmaxf(ag.y + xv.y * sl + b1[c + 1], 0.0f);
    v2f st; st.x = v0; st.y = v1;
    *(v2f*)(x1 + base + c) = st;
    float dot = v0 * score_w[c] + v1 * score_w[c + 1];
    #pragma unroll
    for (int off = 16; off > 0; off >>= 1) dot += __shfl_xor(dot, off, 32);
    float gate = tanhf(dot);
    int k = cid[wid];
    atomicAdd(&sums[(size_t)k * 64 + c],     v0 * gate);
    atomicAdd(&sums[(size_t)k * 64 + c + 1], v1 * gate);
}

// A = A*(1-eye) + eye  (overwrite diagonal with 1)
__global__ void set_diag(float* __restrict__ Amat) {
    int k = blockIdx.x * blockDim.x + threadIdx.x;
    if (k < KCLUST) Amat[(size_t)k * KCLUST + k] = 1.0f;
}

// dinvK[k] = rsqrt(sum_j A[k,j]);  one wave per row
__global__ void rowsum_dinvK(const float* __restrict__ Amat, float* __restrict__ dinvK) {
    int wid = (int)((blockIdx.x * blockDim.x + threadIdx.x) >> 5);
    int lane = threadIdx.x & 31;
    if (wid >= KCLUST) return;
    const float* rowp = Amat + (size_t)wid * KCLUST;
    float s = 0.0f;
    for (int j = lane; j < KCLUST; j += 32) s += rowp[j];
    #pragma unroll
    for (int off = 16; off > 0; off >>= 1) s += __shfl_xor(s, off, 32);
    if (lane == 0) dinvK[wid] = rsqrtf(s);
}

// xp = sums / max(counts, 1)
__global__ void pool_finish(const float* __restrict__ sums, const float* __restrict__ counts,
                            float* __restrict__ xp) {
    int i = blockIdx.x * blockDim.x + threadIdx.x;
    if (i >= KCLUST * 64) return;
    xp[i] = sums[i] / fmaxf(counts[i >> 6], 1.0f);
}

// m[k,:] *= dinvK[k]
__global__ void scale_rows(float* __restrict__ m, const float* __restrict__ dinvK) {
    int i = blockIdx.x * blockDim.x + threadIdx.x;
    if (i >= KCLUST * 64) return;
    m[i] *= dinvK[i >> 6];
}

// xp2 = dinvK[:,None]*z + b2   (in place on z)
__global__ void finish_xp2(float* __restrict__ xp2, const float* __restrict__ dinvK,
                           const float* __restrict__ b2) {
    int i = blockIdx.x * blockDim.x + threadIdx.x;
    if (i >= KCLUST * 64) return;
    xp2[i] = dinvK[i >> 6] * xp2[i] + b2[i & 63];
}

// ---------------- final: out[n,:] = x1[n,:] @ W_skip + b_skip + xp2[cid[n],:] ----------------
// 16x64 strip per wave (Kd = N = 64, unrolled by 2), fused gather/bias epilogue.
// M % 16 == 0 (100000).
__global__ void wmma_skip_out(const float* __restrict__ x1, const float* __restrict__ Wsk,
                              const float* __restrict__ bsk, const float* __restrict__ xp2,
                              const int* __restrict__ cid, float* __restrict__ out, int M) {
    int mt   = (int)((blockIdx.x * blockDim.x + threadIdx.x) >> 5);
    int lane = threadIdx.x & 31;
    if (mt * 16 >= M) return;
    int row   = lane & 15;
    int khalf = lane >> 4;
    const float* ap = x1 + (size_t)(mt * 16 + row) * 64 + 2 * khalf;
    const float* bp = Wsk + (size_t)(2 * khalf) * 64 + row;
    v8f acc0 = {}, acc1 = {}, acc2 = {}, acc3 = {};
    #pragma unroll 2
    for (int k0 = 0; k0 < 64; k0 += 4) {
        v2f a = *(const v2f*)ap;
        v2f b0, b1, b2, b3;
        b0.x = bp[0];       b0.y = bp[64];
        b1.x = bp[16];      b1.y = bp[80];
        b2.x = bp[32];      b2.y = bp[96];
        b3.x = bp[48];      b3.y = bp[112];
        acc0 = __builtin_amdgcn_wmma_f32_16x16x4_f32(false, a, false, b0, (short)0, acc0, false, false);
        acc1 = __builtin_amdgcn_wmma_f32_16x16x4_f32(false, a, false, b1, (short)0, acc1, false, false);
        acc2 = __builtin_amdgcn_wmma_f32_16x16x4_f32(false, a, false, b2, (short)0, acc2, false, false);
        acc3 = __builtin_amdgcn_wmma_f32_16x16x4_f32(false, a, false, b3, (short)0, acc3, false, false);
        ap += 4;
        bp += 256;
    }
    float bias0 = bsk[row];
    float bias1 = bsk[row + 16];
    float bias2 = bsk[row + 32];
    float bias3 = bsk[row + 48];
    int r0 = mt * 16 + 8 * khalf;
    #pragma unroll
    for (int v = 0; v < 8; ++v) {
        int r = r0 + v;
        const float* up = xp2 + (size_t)cid[r] * 64 + row;
        float* op = out + (size_t)r * 64 + row;
        op[0]  = acc0[v] + bias0 + up[0];
        op[16] = acc1[v] + bias1 + up[16];
        op[32] = acc2[v] + bias2 + up[32];
        op[48] = acc3[v] + bias3 + up[48];
    }
}

extern "C" void kernel_launch(void* const* d_in, const int* in_sizes, int n_in,
                              void* d_out, int out_size, void* d_ws, size_t ws_size,
                              hipStream_t stream) {
    const float* x       = (const float*)d_in[0];
    const int*   eidx    = (const int*)d_in[1];
    const int*   cid     = (const int*)d_in[2];
    const float* W1      = (const float*)d_in[3];
    const float* b1      = (const float*)d_in[4];
    const float* score_w = (const float*)d_in[5];
    const float* W2      = (const float*)d_in[6];
    const float* b2      = (const float*)d_in[7];
    const float* W_skip  = (const float*)d_in[8];
    const float* b_skip  = (const float*)d_in[9];
    float* out = (float*)d_out;

    const int Nn = in_sizes[0] / 128;   // 100000
    const int E  = in_sizes[1] / 2;     // 3200000
    const int K  = KCLUST;              // 1000
    const int* src = eidx;
    const int* dst = eidx + E;

    // workspace layout (floats); everything zeroed up front so padded GEMM
    // rows (KCLUST..KPAD-1) contribute zeros with no in-kernel guards.
    float* ws = (float*)d_ws;
    size_t off = 0;
    float* agg    = ws + off; off += (size_t)Nn * 64;    // edge-aggregation accumulator
    float* dinv   = ws + off; off += (size_t)Nn;         // deg -> dinv (in place)
    float* Amat   = ws + off; off += (size_t)KPAD * K;   // cluster adjacency (padded rows)
    float* sums   = ws + off; off += (size_t)K * 64;     // pooled sums
    float* counts = ws + off; off += (size_t)K;          // cluster counts
    float* xw1    = ws + off; off += (size_t)Nn * 64;    // x @ W1
    float* x1b    = ws + off; off += (size_t)Nn * 64;    // relu'd hidden
    float* xp     = ws + off; off += (size_t)KPAD * 64;  // pooled features (padded)
    float* dinvK  = ws + off; off += (size_t)K;          // cluster rsqrt(deg)
    float* xw2    = ws + off; off += (size_t)KPAD * 64;  // xp @ W2, then scaled (padded)
    float* xp2    = ws + off; off += (size_t)KPAD * 64;  // cluster output (padded)
    size_t totalN = off;

    // 1. zero whole workspace (pads + accumulators)
    zero_f32<<<4096, 256, 0, stream>>>(ws, totalN);
    // 2. xw1 = x @ W1   (WMMA fp32, M=Nn, Kd=128)
    {
        int waves = Nn / 16;
        wmma_gemm_n64<<<(waves + 7) / 8, 256, 0, stream>>>(x, W1, xw1, Nn, 128);
    }
    // 3. degree + cluster adjacency histograms
    edge_deg_A<<<(E + 255) / 256, 256, 0, stream>>>(src, dst, cid, dinv, Amat, E);
    // 4. dinv + cluster counts
    node_prep<<<(Nn + 255) / 256, 256, 0, stream>>>(dinv, cid, counts, Nn);
    // 5. edge scatter (one wave per edge)
    scatter_edges<<<(E * 32 + 255) / 256, 256, 0, stream>>>(src, dst, dinv, xw1, agg, E);
    // 6. x1 / gate / pooled sums (one wave per node)
    x1_gate<<<(Nn * 32 + 255) / 256, 256, 0, stream>>>(agg, xw1, dinv, b1, score_w, cid,
                                                       x1b, sums, Nn);
    // 7. adjacency diagonal -> 1
    set_diag<<<(K + 255) / 256, 256, 0, stream>>>(Amat);
    // 8. cluster-graph degree norms
    rowsum_dinvK<<<(K * 32 + 255) / 256, 256, 0, stream>>>(Amat, dinvK);
    // 9. mean pool
    pool_finish<<<(K * 64 + 255) / 256, 256, 0, stream>>>(sums, counts, xp);
    // 10. xw2 = xp @ W2   (padded M=KPAD, Kd=64)
    {
        int waves = KPAD / 16;
        wmma_gemm_n64<<<(waves * 32 + 255) / 256, 256, 0, stream>>>(xp, W2, xw2, KPAD, 64);
    }
    // 11. xw2[k,:] *= dinvK[k]
    scale_rows<<<(K * 64 + 255) / 256, 256, 0, stream>>>(xw2, dinvK);
    // 12. xp2 = Amat @ xw2   (padded M=KPAD, Kd=K)
    {
        int waves = KPAD / 16;
        wmma_gemm_n64<<<(waves * 32 + 255) / 256, 256, 0, stream>>>(Amat, xw2, xp2, KPAD, K);
    }
    // 13. xp2 = dinvK*xp2 + b2
    finish_xp2<<<(K * 64 + 255) / 256, 256, 0, stream>>>(xp2, dinvK, b2);
    // 14. out = x1 @ W_skip + b_skip + xp2[cid]   (fused WMMA epilogue)
    {
        int waves = Nn / 16;
        wmma_skip_out<<<(waves + 7) / 8, 256, 0, stream>>>(x1b, W_skip, b_skip, xp2, cid,
                                                           out, Nn);
    }
}